// MaxDilatedConv2dLayer_70763881169297
// MI455X (gfx1250) — compile-verified
//
#include <hip/hip_runtime.h>
#include <hip/hip_bf16.h>
#include <stdint.h>

// Problem constants (from reference setup_inputs)
#define BATCH 8
#define ICN   32
#define OCN   32
#define KS    3
#define HH    68
#define WW    68
#define DIL   2
#define OHN   64
#define OWN   64

#define ROWS  4                 // output rows per block
#define XROWS (ROWS + 2 * DIL)  // 8 input rows (r0 .. r0+7)
#define OWT   32                // output cols per block
#define XCOLS (OWT + 2 * DIL)   // 36 input cols (ow0 .. ow0+35)
#define ICCH  8                 // input channels per LDS chunk
#define NCHUNK (ICN / ICCH)     // 4
#define NTHREADS 256
#define OCPT  4                 // oc per thread

typedef unsigned int v4u  __attribute__((ext_vector_type(4)));
typedef int          v8i_t __attribute__((ext_vector_type(8)));
typedef int          v4i_t __attribute__((ext_vector_type(4)));

// Build a TDM descriptor (D#) and issue tensor_load_to_lds for a 3D tile:
//   tile = tile0 (contiguous, dim0) x tile1 (dim1, stride s0) x tile2 (dim2, stride s1)
// data_size = 4 bytes. Strides/dims in element units per ISA 8.1/8.4.
__device__ __forceinline__ void tdm_load_3d_f32(uint32_t lds_off, const void* gptr,
                                                uint32_t td0, uint32_t td1, uint32_t td2,
                                                uint32_t tile0, uint32_t tile1, uint32_t tile2,
                                                uint64_t s0, uint64_t s1) {
  uint64_t ga = (uint64_t)(uintptr_t)gptr;
  v4u g0;
  g0.x = 1u;                                                // count=1, user descriptor
  g0.y = lds_off;                                           // lds_addr (bytes)
  g0.z = (uint32_t)ga;                                      // global_addr[31:0]
  g0.w = (uint32_t)((ga >> 32) & 0x1FFFFFFu) | (2u << 30);  // global_addr[56:32] | type=2
  v8i_t g1;
  g1[0] = (int)(2u << 16);                                  // workgroup_mask=0, data_size=2 (4B)
  g1[1] = (int)((td0 & 0xFFFFu) << 16);                     // tensor_dim0[15:0] at bits 63:48
  g1[2] = (int)((td0 >> 16) | ((td1 & 0xFFFFu) << 16));     // td0[31:16] | td1[15:0]
  g1[3] = (int)((td1 >> 16) | (tile0 << 16));               // td1[31:16] | tile_dim0
  g1[4] = (int)(tile1 | (tile2 << 16));                     // tile_dim1 | tile_dim2
  g1[5] = (int)(uint32_t)(s0 & 0xFFFFFFFFu);                // dim0_stride[31:0]
  g1[6] = (int)((uint32_t)((s0 >> 32) & 0xFFFFu) |
                ((uint32_t)(s1 & 0xFFFFu) << 16));          // dim0_stride[47:32] | dim1_stride[15:0]
  g1[7] = (int)(uint32_t)((s1 >> 16) & 0xFFFFFFFFu);        // dim1_stride[47:16]
  v4i_t g2;
  g2[0] = (int)td2;  // tensor_dim2
  g2[1] = 0;         // tensor_dim3 (unused)
  g2[2] = 0;         // tensor_dim2_stride[31:0] (unused)
  g2[3] = 0;         // dim2_stride hi | tile_dim3=0
  v4i_t g3 = {0, 0, 0, 0};
  v8i_t g4 = {0, 0, 0, 0, 0, 0, 0, 0};  // extra group on 6-arg toolchain; zero-filled
  __builtin_amdgcn_tensor_load_to_lds(g0, g1, g2, g3, g4, 0);
}

__global__ void __launch_bounds__(NTHREADS)
MaxDilatedConv2d_kernel(const float* __restrict__ x,
                        const float* __restrict__ w,
                        float* __restrict__ out) {
  // Double-buffered x chunk (8 ic x 8 rows x 36 cols) + transposed weights
  __shared__ __align__(16) float xs[2][ICCH * XROWS * XCOLS];    // 2 x 9,216 B
  __shared__ __align__(16) float wt[ICN * KS * KS * OCN];        // 36,864 B

  const int tid = (int)threadIdx.x;
  const int nRowT = OHN / ROWS;                   // 16 row tiles
  const int nColT = OWN / OWT;                    // 2 col tiles
  int bid = (int)blockIdx.x;
  const int ct  = bid % nColT;  bid /= nColT;     // col tile
  const int rt  = bid % nRowT;  bid /= nRowT;     // row tile
  const int b   = bid;                            // batch
  const int r0  = rt * ROWS;
  const int c0  = ct * OWT;
  const int owl = tid & (OWT - 1);                // local out col, stride-1 in wave
  const int ocg = tid >> 5;                       // oc group (0..7), uniform per wave

  // Cooperative transpose of weights into LDS: wt[(ic*9 + kh*3 + kw)*32 + oc]
  // so per-term weight reads are one broadcast ds_load_b128 (4 oc).
  for (int i = tid; i < OCN * ICN * KS * KS; i += NTHREADS) {
    int oc = i / (ICN * KS * KS);
    int r  = i - oc * (ICN * KS * KS);            // ic*9 + kh*3 + kw
    wt[r * OCN + oc] = w[i];
  }

  float acc[ROWS][OCPT];
#pragma unroll
  for (int p = 0; p < ROWS; ++p)
#pragma unroll
    for (int c = 0; c < OCPT; ++c)
      acc[p][c] = -__builtin_huge_valf();

  const uint32_t xs_lds0 = (uint32_t)(uintptr_t)(&xs[0][0]);
  const uint32_t xs_lds1 = (uint32_t)(uintptr_t)(&xs[1][0]);
  const float* gbase = x + (size_t)(b * ICN) * (HH * WW) + (size_t)(r0 * WW + c0);

  // Prologue: wave 0 kicks off TDM for chunk 0 into buffer 0.
  if (tid < 32) {
    tdm_load_3d_f32(xs_lds0, gbase,
                    XCOLS, HH, ICN,          // tensor dims (in-bounds tile anyway)
                    XCOLS, XROWS, ICCH,      // tile: 36 x 8 x 8
                    (uint64_t)WW, (uint64_t)HH * WW);
  }

#pragma unroll 1
  for (int ch = 0; ch < NCHUNK; ++ch) {
    if (tid < 32) {
      __builtin_amdgcn_s_wait_tensorcnt(0);  // TDM(ch) complete
    }
    __syncthreads();                         // buf[ch&1] (and wt on first pass) visible
    if ((tid < 32) && (ch + 1 < NCHUNK)) {   // overlap: prefetch next chunk via TDM
      const float* gsrc = gbase + (size_t)((ch + 1) * ICCH) * (HH * WW);
      tdm_load_3d_f32((ch & 1) ? xs_lds0 : xs_lds1, gsrc,
                      XCOLS, HH, ICN,
                      XCOLS, XROWS, ICCH,
                      (uint64_t)WW, (uint64_t)HH * WW);
    }

    const float* xbuf = &xs[ch & 1][0];
#pragma unroll 1
    for (int i8 = 0; i8 < ICCH; ++i8) {
      const int ic = ch * ICCH + i8;
      const float* xpl = xbuf + i8 * (XROWS * XCOLS);
#pragma unroll
      for (int kh = 0; kh < KS; ++kh) {
#pragma unroll
        for (int kw = 0; kw < KS; ++kw) {
          const float* wp = &wt[(ic * 9 + kh * 3 + kw) * OCN + ocg * OCPT];
          float4 w0 = *(const float4*)(wp);  // one broadcast ds_load_b128
          float wv[OCPT] = {w0.x, w0.y, w0.z, w0.w};
#pragma unroll
          for (int p = 0; p < ROWS; ++p) {
            float xv = xpl[(p + kh * DIL) * XCOLS + (owl + kw * DIL)];
#pragma unroll
            for (int c = 0; c < OCPT; ++c)
              acc[p][c] = fmaxf(acc[p][c], xv * wv[c]);
          }
        }
      }
    }
    // no trailing barrier needed: next TDM writes the *other* buffer, and the
    // barrier at the top of the next iteration orders readers vs that write.
  }

  // Store: out[b][oc][oh][ow], coalesced along ow
#pragma unroll
  for (int p = 0; p < ROWS; ++p) {
#pragma unroll
    for (int c = 0; c < OCPT; ++c) {
      int oc = ocg * OCPT + c;
      out[(((size_t)b * OCN + oc) * OHN + (r0 + p)) * OWN + (c0 + owl)] = acc[p][c];
    }
  }
}

extern "C" void kernel_launch(void* const* d_in, const int* in_sizes, int n_in,
                              void* d_out, int out_size, void* d_ws, size_t ws_size,
                              hipStream_t stream) {
  (void)in_sizes; (void)n_in; (void)d_ws; (void)ws_size; (void)out_size;
  const float* x = (const float*)d_in[0];
  const float* w = (const float*)d_in[1];
  float* out = (float*)d_out;
  dim3 grid(BATCH * (OHN / ROWS) * (OWN / OWT));  // 256 blocks
  dim3 block(NTHREADS);                           // 8 wave32 per block
  MaxDilatedConv2d_kernel<<<grid, block, 0, stream>>>(x, w, out);
}